// CayleyConv_10342281249235
// MI455X (gfx1250) — compile-verified
//
#include <hip/hip_runtime.h>
#include <hip/hip_bf16.h>

// ---------------------------------------------------------------------------
// CayleyNet graph conv for gfx1250 (MI455X).
//  - CSR built on-device (count / 2-level scan / fill): gather SpMM, no float atomics
//  - fused complex Jacobi sweep kernels (wave-per-node, wave32)
//  - dense 64x64 linear layers via v_wmma_f32_16x16x4_f32 (exact FP32 WMMA),
//    MODE as template param so the unrolled K-loop has no per-chunk branches
// ---------------------------------------------------------------------------

typedef __attribute__((ext_vector_type(2))) float v2f;
typedef __attribute__((ext_vector_type(8))) float v8f;

#define FDIM 64
#define KJAC 10

// ------------------------------ CSR build ----------------------------------

__global__ void init0_k(float* deg, int* cnt, int* fill, int N) {
    int i = blockIdx.x * blockDim.x + threadIdx.x;
    if (i < N) { deg[i] = 0.0f; cnt[i] = 0; fill[i] = 0; }
}

__global__ void count_k(const int* __restrict__ row, const float* __restrict__ ew,
                        float* deg, int* cnt, int E) {
    int e = blockIdx.x * blockDim.x + threadIdx.x;
    if (e >= E) return;
    int r = row[e];
    atomicAdd(&deg[r], ew[e]);
    atomicAdd(&cnt[r], 1);
}

__global__ void dis_k(const float* __restrict__ deg, float* dis, int N) {
    int i = blockIdx.x * blockDim.x + threadIdx.x;
    if (i < N) {
        float d = deg[i];
        dis[i] = (d > 0.0f) ? rsqrtf(fmaxf(d, 1e-30f)) : 0.0f;
    }
}

// exclusive scan over M = N+1 virtual elements (cnt[N] := 0) -> ptr[0..N]
__global__ void scan1_k(const int* __restrict__ cnt, int* ptr, int* bsum, int N, int M) {
    __shared__ int sh[256];
    int t = threadIdx.x;
    int i = blockIdx.x * 256 + t;
    int c = (i < N) ? cnt[i] : 0;
    sh[t] = c;
    __syncthreads();
    for (int off = 1; off < 256; off <<= 1) {
        int v = (t >= off) ? sh[t - off] : 0;
        __syncthreads();
        sh[t] += v;
        __syncthreads();
    }
    if (i < M) ptr[i] = sh[t] - c;          // exclusive within block
    if (t == 255) bsum[blockIdx.x] = sh[255];
}

__global__ void scan2_k(const int* __restrict__ bsum, int* boff, int NB) {
    __shared__ int sh[512];
    int t = threadIdx.x;
    int c = (t < NB) ? bsum[t] : 0;
    sh[t] = c;
    __syncthreads();
    for (int off = 1; off < 512; off <<= 1) {
        int v = (t >= off) ? sh[t - off] : 0;
        __syncthreads();
        sh[t] += v;
        __syncthreads();
    }
    boff[t] = sh[t] - c;                    // exclusive block offsets
}

__global__ void scan3_k(int* ptr, const int* __restrict__ boff, int M) {
    int i = blockIdx.x * 256 + threadIdx.x;
    if (i < M) ptr[i] += boff[i >> 8];
}

__global__ void fill_k(const int* __restrict__ row, const int* __restrict__ col,
                       const float* __restrict__ ew, const float* __restrict__ h,
                       const float* __restrict__ dis, const int* __restrict__ ptr,
                       int* fill, int* ccol, float* cval, int E) {
    int e = blockIdx.x * blockDim.x + threadIdx.x;
    if (e >= E) return;
    int r = row[e], c = col[e];
    int pos = ptr[r] + atomicAdd(&fill[r], 1);
    ccol[pos] = c;
    // off-diag of hL:  h * (-dis[r] * w * dis[c])
    cval[pos] = -h[0] * dis[r] * ew[e] * dis[c];
}

__global__ void inity_k(const float* __restrict__ x, float* yre, float* yim, int total) {
    int i = blockIdx.x * blockDim.x + threadIdx.x;
    if (i < total) { yre[i] = x[i]; yim[i] = 0.0f; }
}

// ----------------------- fused SpMM kernels (wave/node) --------------------
// b = A_off@y + diag_B*y ; d = dinv*b ; z0 = b
__global__ void bd_k(const int* __restrict__ ptr, const int* __restrict__ ccol,
                     const float* __restrict__ cval,
                     const float* __restrict__ yre, const float* __restrict__ yim,
                     const float* __restrict__ h,
                     float* zre, float* zim, float* dre, float* dim_, int N) {
    int node = blockIdx.x * 8 + (threadIdx.x >> 5);
    if (node >= N) return;
    int lane = threadIdx.x & 31;
    int f0 = lane, f1 = lane + 32;
    int beg = ptr[node], end = ptr[node + 1];
    float sr0 = 0, sr1 = 0, si0 = 0, si1 = 0;
    for (int e = beg; e < end; ++e) {
        int c = ccol[e];
        float v = cval[e];
        const float* zr = yre + (size_t)c * FDIM;
        const float* zi = yim + (size_t)c * FDIM;
        sr0 = fmaf(v, zr[f0], sr0); sr1 = fmaf(v, zr[f1], sr1);
        si0 = fmaf(v, zi[f0], si0); si1 = fmaf(v, zi[f1], si1);
    }
    float hh = h[0];
    float inv = 1.0f / (hh * hh + 1.0f);
    size_t base = (size_t)node * FDIM;
    float yr0 = yre[base + f0], yr1 = yre[base + f1];
    float yi0 = yim[base + f0], yi1 = yim[base + f1];
    // diag_B = h - i :  (h*yr + yi) + i*(h*yi - yr)
    float br0 = sr0 + hh * yr0 + yi0, bi0 = si0 + hh * yi0 - yr0;
    float br1 = sr1 + hh * yr1 + yi1, bi1 = si1 + hh * yi1 - yr1;
    zre[base + f0] = br0; zre[base + f1] = br1;
    zim[base + f0] = bi0; zim[base + f1] = bi1;
    // dinv = (h - i)/(h^2+1)
    dre[base + f0] = (hh * br0 + bi0) * inv; dre[base + f1] = (hh * br1 + bi1) * inv;
    dim_[base + f0] = (hh * bi0 - br0) * inv; dim_[base + f1] = (hh * bi1 - br1) * inv;
}

// z_out = d - dinv * (A_off @ z_in)
__global__ void jacobi_k(const int* __restrict__ ptr, const int* __restrict__ ccol,
                         const float* __restrict__ cval,
                         const float* __restrict__ zin_re, const float* __restrict__ zin_im,
                         const float* __restrict__ dre, const float* __restrict__ dim_,
                         const float* __restrict__ h,
                         float* zout_re, float* zout_im, int N) {
    int node = blockIdx.x * 8 + (threadIdx.x >> 5);
    if (node >= N) return;
    int lane = threadIdx.x & 31;
    int f0 = lane, f1 = lane + 32;
    int beg = ptr[node], end = ptr[node + 1];
    float sr0 = 0, sr1 = 0, si0 = 0, si1 = 0;
    for (int e = beg; e < end; ++e) {
        int c = ccol[e];
        float v = cval[e];
        if (e + 1 < end) {
            const float* nxt = zin_re + (size_t)ccol[e + 1] * FDIM;
            __builtin_prefetch(nxt, 0, 3);   // global_prefetch_b8
        }
        const float* zr = zin_re + (size_t)c * FDIM;
        const float* zi = zin_im + (size_t)c * FDIM;
        sr0 = fmaf(v, zr[f0], sr0); sr1 = fmaf(v, zr[f1], sr1);
        si0 = fmaf(v, zi[f0], si0); si1 = fmaf(v, zi[f1], si1);
    }
    float hh = h[0];
    float inv = 1.0f / (hh * hh + 1.0f);
    float tr0 = (hh * sr0 + si0) * inv, ti0 = (hh * si0 - sr0) * inv;
    float tr1 = (hh * sr1 + si1) * inv, ti1 = (hh * si1 - sr1) * inv;
    size_t base = (size_t)node * FDIM;
    zout_re[base + f0] = dre[base + f0] - tr0;
    zout_re[base + f1] = dre[base + f1] - tr1;
    zout_im[base + f0] = dim_[base + f0] - ti0;
    zout_im[base + f1] = dim_[base + f1] - ti1;
}

// --------------------------- WMMA linear layers ----------------------------
// V_WMMA_F32_16X16X4_F32 fragment layout (ISA 7.12.2, 32-bit A 16x4):
//   A: lane l+16g holds row m=l, elements j=0,1 -> K = 2g + j
//   B: lane l+16g holds col n=l, elements j=0,1 -> K = 2g + j   (symmetric)
//   C/D: VGPR r <-> row m0 + r + 8g, column n0 + l
//
// MODE 0: out  = Are @ Wa^T
// MODE 1: out += 2*(Are @ Wa^T - Aim @ Wb^T)
template <int MODE>
__global__ void wmma_linear_k(const float* __restrict__ Are, const float* __restrict__ Aim,
                              const float* __restrict__ Wa,  const float* __restrict__ Wb,
                              float* __restrict__ out, int N) {
    int wave = threadIdx.x >> 5;              // 8 waves per block
    int lane = threadIdx.x & 31;
    int l = lane & 15, g = lane >> 4;
    int mtile = blockIdx.x * 2 + (wave >> 2); // 16-node tile
    int ntile = wave & 3;                     // 16-outfeat tile
    int m0 = mtile * 16, n0 = ntile * 16;
    if (m0 >= N) return;                      // uniform per wave -> EXEC stays full
    int mrow = m0 + l; if (mrow > N - 1) mrow = N - 1;   // clamp keeps all lanes live
    const float2* arow  = (const float2*)(Are + (size_t)mrow * FDIM);
    const float2* wrow  = (const float2*)(Wa + (size_t)(n0 + l) * FDIM);
    const float2* airow = MODE ? (const float2*)(Aim + (size_t)mrow * FDIM) : nullptr;
    const float2* wbrow = MODE ? (const float2*)(Wb + (size_t)(n0 + l) * FDIM) : nullptr;
    v8f c = {};
    #pragma unroll
    for (int kb = 0; kb < FDIM; kb += 4) {
        int kp = (kb >> 1) + g;               // float2 index of K = kb + 2g
        float2 av = arow[kp];
        float2 bv = wrow[kp];
        v2f a; a.x = av.x; a.y = av.y;
        v2f b; b.x = bv.x; b.y = bv.y;
        c = __builtin_amdgcn_wmma_f32_16x16x4_f32(false, a, false, b,
                                                  (short)0, c, false, false);
        if (MODE) {
            float2 av2 = airow[kp];
            float2 bv2 = wbrow[kp];
            v2f a2; a2.x = -av2.x; a2.y = -av2.y;   // fold the minus sign here
            v2f b2; b2.x = bv2.x;  b2.y = bv2.y;
            c = __builtin_amdgcn_wmma_f32_16x16x4_f32(false, a2, false, b2,
                                                      (short)0, c, false, false);
        }
    }
    // C/D layout: VGPR r <-> row m0 + r + 8g, column n0 + l
    #pragma unroll
    for (int r = 0; r < 8; ++r) {
        int rowi = m0 + r + 8 * g;
        if (rowi < N) {
            size_t idx = (size_t)rowi * FDIM + n0 + l;
            if (MODE) out[idx] += 2.0f * c[r];
            else      out[idx] = c[r];
        }
    }
}

// ------------------------------- launcher ----------------------------------

extern "C" void kernel_launch(void* const* d_in, const int* in_sizes, int n_in,
                              void* d_out, int out_size, void* d_ws, size_t ws_size,
                              hipStream_t stream) {
    (void)n_in; (void)out_size; (void)ws_size;
    const float* x   = (const float*)d_in[0];
    const int*   ei  = (const int*)d_in[1];
    const float* ew  = (const float*)d_in[2];
    const float* h   = (const float*)d_in[3];
    const float* W0  = (const float*)d_in[4];
    const float* Wre = (const float*)d_in[5];
    const float* Wim = (const float*)d_in[6];
    float* out = (float*)d_out;

    const int N = in_sizes[0] / FDIM;
    const int E = in_sizes[2];
    const int R = in_sizes[5] / (FDIM * FDIM);
    const int* row = ei;
    const int* col = ei + E;

    // workspace carve-up (256B aligned)
    char* w = (char*)d_ws;
    auto alloc = [&](size_t bytes) -> char* {
        char* p = w;
        w += (bytes + 255) & ~(size_t)255;
        return p;
    };
    float* deg  = (float*)alloc((size_t)N * 4);
    float* dis  = (float*)alloc((size_t)N * 4);
    int*   cnt  = (int*)alloc((size_t)N * 4);
    int*   fill = (int*)alloc((size_t)N * 4);
    int*   ptr  = (int*)alloc((size_t)(N + 1) * 4);
    int*   bsum = (int*)alloc(512 * 4);
    int*   boff = (int*)alloc(513 * 4);
    int*   ccol = (int*)alloc((size_t)E * 4);
    float* cval = (float*)alloc((size_t)E * 4);
    size_t nf = (size_t)N * FDIM * 4;
    float* cur_re = (float*)alloc(nf); float* cur_im = (float*)alloc(nf);
    float* za_re  = (float*)alloc(nf); float* za_im  = (float*)alloc(nf);
    float* dd_re  = (float*)alloc(nf); float* dd_im  = (float*)alloc(nf);

    const int TB = 256;
    // CSR build
    init0_k<<<(N + TB - 1) / TB, TB, 0, stream>>>(deg, cnt, fill, N);
    count_k<<<(E + TB - 1) / TB, TB, 0, stream>>>(row, ew, deg, cnt, E);
    dis_k<<<(N + TB - 1) / TB, TB, 0, stream>>>(deg, dis, N);
    int M = N + 1;
    int NB = (M + 255) / 256;                 // <= 512 for N <= 131071
    scan1_k<<<NB, 256, 0, stream>>>(cnt, ptr, bsum, N, M);
    scan2_k<<<1, 512, 0, stream>>>(bsum, boff, NB);
    scan3_k<<<NB, 256, 0, stream>>>(ptr, boff, M);
    fill_k<<<(E + TB - 1) / TB, TB, 0, stream>>>(row, col, ew, h, dis, ptr, fill,
                                                 ccol, cval, E);
    // y0 = x + 0i ; out = x @ W0^T
    inity_k<<<((N * FDIM) + TB - 1) / TB, TB, 0, stream>>>(x, cur_re, cur_im, N * FDIM);
    wmma_linear_k<0><<<(N + 31) / 32, 256, 0, stream>>>(x, nullptr, W0, nullptr, out, N);

    int gspmm = (N + 7) / 8;
    for (int i = 0; i < R; ++i) {
        bd_k<<<gspmm, 256, 0, stream>>>(ptr, ccol, cval, cur_re, cur_im, h,
                                        za_re, za_im, dd_re, dd_im, N);
        float *a_re = za_re, *a_im = za_im;   // holds z_t
        float *b_re = cur_re, *b_im = cur_im; // scratch (y is dead now)
        for (int t = 0; t < KJAC; ++t) {
            jacobi_k<<<gspmm, 256, 0, stream>>>(ptr, ccol, cval, a_re, a_im,
                                                dd_re, dd_im, h, b_re, b_im, N);
            float* s;
            s = a_re; a_re = b_re; b_re = s;
            s = a_im; a_im = b_im; b_im = s;
        }
        // a_* now holds y_{i+1}; accumulate 2*(Re@Wre^T - Im@Wim^T)
        wmma_linear_k<1><<<(N + 31) / 32, 256, 0, stream>>>(
            a_re, a_im, Wre + (size_t)i * FDIM * FDIM, Wim + (size_t)i * FDIM * FDIM,
            out, N);
        // next y = a ; recycle b as next zA
        cur_re = a_re; cur_im = a_im;
        za_re = b_re;  za_im = b_im;
    }
}